// PassageRankerNoun_8830452761093
// MI455X (gfx1250) — compile-verified
//
#include <hip/hip_runtime.h>
#include <math.h>

// Problem constants (from reference)
#define L_ 8
#define P_ 8
#define S_ 128
#define B_ 16
#define W_ 768
#define M_ 64

typedef float v2f __attribute__((ext_vector_type(2)));
typedef float v8f __attribute__((ext_vector_type(8)));

// ---------------------------------------------------------------------------
// Kernel 1: single streaming pass over embeds (403 MB -> the whole cost).
// grid = L*P*B = 1024 blocks, block = 192 threads (6 waves).
// Thread t owns columns w = 4t..4t+3 (one float4). Per s-row:
//   - update running max (for max-pool over S)
//   - acc += w1[s] * e   (fused steps 1+2: score = sum_w w0[w]*acc[w] + bias)
// One block reduction at the end -> sigmoid(score).
// ---------------------------------------------------------------------------
__global__ void __launch_bounds__(192)
k1_stream_fuse(const float* __restrict__ embeds,
               const float* __restrict__ w0,
               const float* __restrict__ b0,
               const float* __restrict__ w1,
               const float* __restrict__ b1,
               float* __restrict__ mp,      // [L*P, B, W]
               float* __restrict__ sig)     // [L*P, B]
{
    const int lp = blockIdx.x / B_;         // l*P + p
    const int b  = blockIdx.x % B_;
    const int t  = threadIdx.x;             // 0..191

    const float* base = embeds + (size_t)lp * (S_ * B_ * W_) + (size_t)b * W_;

    const float4 w0v = ((const float4*)w0)[t];

    float mx0 = -INFINITY, mx1 = -INFINITY, mx2 = -INFINITY, mx3 = -INFINITY;
    float a0 = 0.f, a1 = 0.f, a2 = 0.f, a3 = 0.f;

    #pragma unroll 4
    for (int s = 0; s < S_; ++s) {
        const float4 e = *(const float4*)(base + (size_t)s * (B_ * W_) + 4 * t);
        const float w1s = w1[s];
        mx0 = fmaxf(mx0, e.x); mx1 = fmaxf(mx1, e.y);
        mx2 = fmaxf(mx2, e.z); mx3 = fmaxf(mx3, e.w);
        a0 = fmaf(w1s, e.x, a0); a1 = fmaf(w1s, e.y, a1);
        a2 = fmaf(w1s, e.z, a2); a3 = fmaf(w1s, e.w, a3);
    }

    // write max-pool row
    float4* mpo = (float4*)(mp + (size_t)lp * (B_ * W_) + (size_t)b * W_);
    float4 mxv; mxv.x = mx0; mxv.y = mx1; mxv.z = mx2; mxv.w = mx3;
    mpo[t] = mxv;

    // fused score contribution: w0[w] * (sum_s w1[s]*e[s,w])
    float v = a0 * w0v.x + a1 * w0v.y + a2 * w0v.z + a3 * w0v.w;

    // wave32 reduction
    #pragma unroll
    for (int off = 16; off > 0; off >>= 1)
        v += __shfl_down(v, off, 32);

    __shared__ float red[6];
    if ((t & 31) == 0) red[t >> 5] = v;
    __syncthreads();

    if (t == 0) {
        float score = red[0] + red[1] + red[2] + red[3] + red[4] + red[5];
        // bias terms: score += b0 * sum_s(w1[s]) + b1
        const float bb0 = b0[0];
        if (bb0 != 0.f) {
            float sw1 = 0.f;
            for (int s = 0; s < S_; ++s) sw1 += w1[s];
            score = fmaf(bb0, sw1, score);
        }
        score += b1[0];
        sig[blockIdx.x] = 1.f / (1.f + expf(-score));
    }
}

// ---------------------------------------------------------------------------
// Kernel 2: softmax over P per (l,b), then sem_sum[l,b,w] = sum_p mp * weight.
// grid = L*B = 128 blocks, block = 256 threads, 3 w's per thread.
// ---------------------------------------------------------------------------
__global__ void __launch_bounds__(256)
k2_softmax_pool(const float* __restrict__ mp,    // [L*P, B, W]
                const float* __restrict__ sig,   // [L*P, B]
                float* __restrict__ sem)         // [L, B, W]
{
    const int l = blockIdx.x / B_;
    const int b = blockIdx.x % B_;
    const int t = threadIdx.x;

    __shared__ float esh[P_];
    if (t < P_) esh[t] = expf(sig[(l * P_ + t) * B_ + b]);
    __syncthreads();

    float denom = 0.f;
    #pragma unroll
    for (int p = 0; p < P_; ++p) denom += esh[p];
    const float inv = 1.f / denom;

    #pragma unroll
    for (int j = 0; j < 3; ++j) {
        const int w = t + 256 * j;
        float acc = 0.f;
        #pragma unroll
        for (int p = 0; p < P_; ++p)
            acc = fmaf(mp[((size_t)(l * P_ + p) * B_ + b) * W_ + w], esh[p], acc);
        sem[((size_t)l * B_ + b) * W_ + w] = acc * inv;
    }
}

// ---------------------------------------------------------------------------
// Kernel 3: out[b,m,w] = sum_l mask[b,l,m] * sem[l,b,w]
// Per b this is maskT[M=64 x L=8] @ sem_b[L=8 x W=768]:
// V_WMMA_F32_16X16X4_F32, two K-steps (K=4 each) per 16x16 output tile.
// One wave (32 threads, EXEC all ones) per tile.
// A layout (16x4 f32, ISA 7.12.2): lane<16 -> K={0,1} in v0/v1; lane>=16 -> K={2,3}.
// B layout (4x16 f32): v0 = row K0 (lanes 0-15) / K2 (lanes 16-31); v1 = K1/K3.
// C/D layout: VGPR r -> row M=r (lanes 0-15), M=r+8 (lanes 16-31), col = lane%16.
// ---------------------------------------------------------------------------
__global__ void __launch_bounds__(32)
k3_wmma_out(const float* __restrict__ mask,   // [B, L, M]
            const float* __restrict__ sem,    // [L, B, W]
            float* __restrict__ out)          // [B, M, W]
{
    const int bid = blockIdx.x;               // b*(4*48) + mt*48 + wt
    const int wt  = bid % 48;
    const int mt  = (bid / 48) % 4;
    const int b   = bid / (48 * 4);

    const int lane = threadIdx.x;
    const int half = lane >> 4;               // 0 or 1
    const int lr   = lane & 15;

    const int mrow = mt * 16 + lr;            // A-matrix row for this lane
    const int wcol = wt * 16 + lr;            // B/D column for this lane

    v8f c = {};
    #pragma unroll
    for (int k0 = 0; k0 < L_; k0 += 4) {
        const int kk = k0 + 2 * half;         // this half-wave's K base
        v2f a, bm;
        a.x  = mask[((size_t)b * L_ + kk    ) * M_ + mrow];
        a.y  = mask[((size_t)b * L_ + kk + 1) * M_ + mrow];
        bm.x = sem[((size_t)(kk    ) * B_ + b) * W_ + wcol];
        bm.y = sem[((size_t)(kk + 1) * B_ + b) * W_ + wcol];
        c = __builtin_amdgcn_wmma_f32_16x16x4_f32(
                /*neg_a=*/false, a, /*neg_b=*/false, bm,
                /*c_mod=*/(short)0, c, /*reuse_a=*/false, /*reuse_b=*/false);
    }

    #pragma unroll
    for (int r = 0; r < 8; ++r) {
        const int row = mt * 16 + r + 8 * half;
        out[((size_t)b * M_ + row) * W_ + wcol] = c[r];
    }
}

// ---------------------------------------------------------------------------
extern "C" void kernel_launch(void* const* d_in, const int* in_sizes, int n_in,
                              void* d_out, int out_size, void* d_ws, size_t ws_size,
                              hipStream_t stream) {
    const float* embeds = (const float*)d_in[0];   // [L,P,S,B,W]
    const float* mask   = (const float*)d_in[1];   // [B,L,M]
    const float* w0     = (const float*)d_in[2];   // [1,W]
    const float* b0     = (const float*)d_in[3];   // [1]
    const float* w1     = (const float*)d_in[4];   // [1,S]
    const float* b1     = (const float*)d_in[5];   // [1]
    float* out = (float*)d_out;                    // [B,M,W]

    // workspace layout (floats)
    float* wsf = (float*)d_ws;
    float* mp  = wsf;                                  // L*P*B*W = 786432
    float* sig = mp + (size_t)L_ * P_ * B_ * W_;       // L*P*B   = 1024
    float* sem = sig + (size_t)L_ * P_ * B_;           // L*B*W   = 98304

    // K1: one streaming pass over embeds (the only bandwidth-significant step)
    k1_stream_fuse<<<L_ * P_ * B_, 192, 0, stream>>>(embeds, w0, b0, w1, b1, mp, sig);

    // K2: softmax over P + weighted pool -> sem_sum
    k2_softmax_pool<<<L_ * B_, 256, 0, stream>>>(mp, sig, sem);

    // K3: WMMA f32 16x16x4 for the final L-contraction
    k3_wmma_out<<<B_ * 4 * 48, 32, 0, stream>>>(mask, sem, out);
}